// RNBlock_41540923687557
// MI455X (gfx1250) — compile-verified
//
#include <hip/hip_runtime.h>

typedef float v2f __attribute__((ext_vector_type(2)));
typedef float v8f __attribute__((ext_vector_type(8)));

#ifndef __has_builtin
#define __has_builtin(x) 0
#endif
#define HAS_WMMA_F32 __has_builtin(__builtin_amdgcn_wmma_f32_16x16x4_f32)

// ---------------------------------------------------------------------------
// Problem constants (B=2, N=512, M=512, D=128, H=256)
// ---------------------------------------------------------------------------
#define BB 2
#define NN 512
#define MM 512
#define DD 128
#define HH 256
#define ROWS (BB * NN)   // 1024
#define MCHUNKS 8        // m split for pool parallelism (64 m per chunk)

// ---------------------------------------------------------------------------
// mask transpose: maskT[b][m][n] = (float)mask[b][n][m]
// ---------------------------------------------------------------------------
__global__ void __launch_bounds__(256) transpose_mask(const int* __restrict__ mask,
                                                      float* __restrict__ maskT) {
    int idx = blockIdx.x * 256 + threadIdx.x;        // 0 .. B*M*N-1, n fastest
    int n = idx & (NN - 1);
    int m = (idx >> 9) & (MM - 1);
    int b = idx >> 18;
    maskT[idx] = (float)mask[(b * NN + n) * MM + m];
}

// cnt[b*N+n] = sum_m mask[b][n][m]
__global__ void __launch_bounds__(256) count_mask(const int* __restrict__ mask,
                                                  float* __restrict__ cnt) {
    int row = blockIdx.x * 256 + threadIdx.x;        // 0..1023
    if (row >= ROWS) return;
    const int* mp = mask + row * MM;
    int s = 0;
    for (int m = 0; m < MM; ++m) s += mp[m];
    cnt[row] = (float)s;
}

// ---------------------------------------------------------------------------
// Generic f32 WMMA GEMM: C[Mrows x Ncols] = A[Mrows x K] * Bm[K x Ncols] + epi
//   MODE 0: none       MODE 1: +bias[col]          MODE 2: relu(acc+bias[col])
//   MODE 3: +res+bias  MODE 4: +res+cnt[row]*bias[col]
//   ATRANS: A stored K-major as A[k*lda + row] (coalesced per-lane loads)
// One 16x16 C tile per wave; 8 waves per workgroup.
// ---------------------------------------------------------------------------
template <int MODE, bool ATRANS>
__global__ void __launch_bounds__(256) wmma_gemm_f32(
    const float* __restrict__ A, int lda,
    const float* __restrict__ Bm, int ldb,
    float* __restrict__ C, int Ncols, int Mrows, int K,
    const float* __restrict__ bias,
    const float* __restrict__ res,
    const float* __restrict__ cnt) {
    const int lane = threadIdx.x & 31;
    const int wave = threadIdx.x >> 5;
    const int ntiles = Ncols >> 4;
    const int gtile = blockIdx.x * 8 + wave;
    if (gtile >= (Mrows >> 4) * ntiles) return;      // wave-uniform (EXEC stays full)
    const int tm = gtile / ntiles;
    const int tn = gtile - tm * ntiles;
    const int hl = lane >> 4;                        // half-wave select
    const int l16 = lane & 15;
    const int m0 = tm * 16, n0 = tn * 16;
    const int col = n0 + l16;

    float accs[8];
#if HAS_WMMA_F32
    v8f acc = {0.f, 0.f, 0.f, 0.f, 0.f, 0.f, 0.f, 0.f};
#pragma unroll 8
    for (int kk = 0; kk < K; kk += 4) {
        // A frag (ISA 16x4 f32 layout): lanes0-15 rows M, v0=K, v1=K+1; lanes16-31 K+2/K+3
        v2f a, b;
        if (ATRANS) {
            const int r = m0 + l16;
            a.x = A[(kk + 2 * hl) * lda + r];
            a.y = A[(kk + 2 * hl + 1) * lda + r];
        } else {
            const float* ap = A + (m0 + l16) * lda + kk + 2 * hl;
            a.x = ap[0];
            a.y = ap[1];
        }
        // B frag (4x16): row K striped across lanes within a VGPR
        b.x = Bm[(kk + 2 * hl) * ldb + col];
        b.y = Bm[(kk + 2 * hl + 1) * ldb + col];
        acc = __builtin_amdgcn_wmma_f32_16x16x4_f32(false, a, false, b,
                                                    (short)0, acc, false, false);
    }
#pragma unroll
    for (int r = 0; r < 8; ++r) accs[r] = acc[r];
#else
    // scalar fallback (same C-tile ownership), used only if builtin missing
#pragma unroll
    for (int r = 0; r < 8; ++r) accs[r] = 0.f;
    for (int k = 0; k < K; ++k) {
        float bv = Bm[k * ldb + col];
#pragma unroll
        for (int r = 0; r < 8; ++r) {
            int row = m0 + r + 8 * hl;
            float av = ATRANS ? A[k * lda + row] : A[row * lda + k];
            accs[r] = fmaf(av, bv, accs[r]);
        }
    }
#endif
    float bcol = 0.f;
    if constexpr (MODE >= 1) bcol = bias[col];
#pragma unroll
    for (int r = 0; r < 8; ++r) {
        const int row = m0 + r + 8 * hl;             // ISA C/D layout: VGPR r -> M=r / r+8
        const int idx = row * Ncols + col;
        float v = accs[r];
        if constexpr (MODE == 1) v += bcol;
        if constexpr (MODE == 2) v = fmaxf(v + bcol, 0.f);
        if constexpr (MODE == 3) v += res[idx] + bcol;
        if constexpr (MODE == 4) v += res[idx] + cnt[row] * bcol;
        C[idx] = v;
    }
}

// ---------------------------------------------------------------------------
// Fused masked relu-pool (the hot kernel):
//   Ppart[ck][b][h][n] = sum_{m in chunk} maskT[b][m][n] *
//                        relu(hy[b][m][h] + hxb[b][n][h])
// lane = n offset (32 n per wave), 32 h channels in registers per lane,
// hy tile staged through 32KB of LDS. Inner body: add + max + fmac per element.
// ---------------------------------------------------------------------------
__global__ void __launch_bounds__(256) pool_kernel(const float* __restrict__ hy,
                                                   const float* __restrict__ hxb,
                                                   const float* __restrict__ maskT,
                                                   float* __restrict__ Ppart) {
    __shared__ float lhy[32 * HH];                   // 32 m-rows of H channels
    const int tid = threadIdx.x;
    const int lane = tid & 31;
    const int wave = tid >> 5;
    const int ntile = blockIdx.x;                    // 0..15
    const int ck = blockIdx.y;                       // 0..MCHUNKS-1
    const int b = blockIdx.z;                        // 0..1
    const int n = ntile * 32 + lane;
    const int h0 = wave * 32;

    float hxr[32], acc[32];
    const float4* hxp = (const float4*)(hxb + (b * NN + n) * HH + h0);
#pragma unroll
    for (int j4 = 0; j4 < 8; ++j4) {
        float4 q = hxp[j4];
        hxr[4 * j4 + 0] = q.x;
        hxr[4 * j4 + 1] = q.y;
        hxr[4 * j4 + 2] = q.z;
        hxr[4 * j4 + 3] = q.w;
    }
#pragma unroll
    for (int j = 0; j < 32; ++j) acc[j] = 0.f;

    for (int s = 0; s < 2; ++s) {                    // 2 sub-tiles of 32 m
        const int m0 = ck * (MM / MCHUNKS) + s * 32;
        __syncthreads();                             // protect lhy reuse
        const float4* src = (const float4*)(hy + (b * MM + m0) * HH);
        float4* dst = (float4*)lhy;
        for (int i = tid; i < 32 * HH / 4; i += 256) dst[i] = src[i];
        __syncthreads();
#pragma unroll 4
        for (int mm = 0; mm < 32; ++mm) {
            const float mf = maskT[(b * MM + m0 + mm) * NN + n]; // coalesced
            const float4* hv = (const float4*)(lhy + mm * HH + h0);
#pragma unroll
            for (int j4 = 0; j4 < 8; ++j4) {
                float4 q = hv[j4];
                acc[4 * j4 + 0] = fmaf(mf, fmaxf(q.x + hxr[4 * j4 + 0], 0.f), acc[4 * j4 + 0]);
                acc[4 * j4 + 1] = fmaf(mf, fmaxf(q.y + hxr[4 * j4 + 1], 0.f), acc[4 * j4 + 1]);
                acc[4 * j4 + 2] = fmaf(mf, fmaxf(q.z + hxr[4 * j4 + 2], 0.f), acc[4 * j4 + 2]);
                acc[4 * j4 + 3] = fmaf(mf, fmaxf(q.w + hxr[4 * j4 + 3], 0.f), acc[4 * j4 + 3]);
            }
        }
    }
#pragma unroll
    for (int j = 0; j < 32; ++j)                     // coalesced over lanes (n fastest)
        Ppart[((ck * BB + b) * HH + h0 + j) * NN + n] = acc[j];
}

// PT[h][b*N+n] = sum_ck Ppart[ck][b][h][n]   (K-major so WMMA A-loads coalesce)
__global__ void __launch_bounds__(256) reduce_P(const float* __restrict__ Ppart,
                                                float* __restrict__ PT) {
    int idx = blockIdx.x * 256 + threadIdx.x;        // 0 .. H*ROWS-1
    int h = idx >> 10;                               // / ROWS
    int rowg = idx & (ROWS - 1);
    int b = rowg >> 9, n = rowg & (NN - 1);
    float s = 0.f;
#pragma unroll
    for (int ck = 0; ck < MCHUNKS; ++ck)
        s += Ppart[((ck * BB + b) * HH + h) * NN + n];
    PT[h * ROWS + rowg] = s;
}

// ---------------------------------------------------------------------------
// LayerNorm over D=128: one row per 128-thread block
// ---------------------------------------------------------------------------
__global__ void __launch_bounds__(128) layernorm_kernel(const float* __restrict__ in,
                                                        const float* __restrict__ g,
                                                        const float* __restrict__ be,
                                                        float* __restrict__ out) {
    __shared__ float red[128];
    const int row = blockIdx.x, t = threadIdx.x;
    float x = in[row * DD + t];
    red[t] = x;
    __syncthreads();
    for (int off = 64; off > 0; off >>= 1) {
        if (t < off) red[t] += red[t + off];
        __syncthreads();
    }
    float mu = red[0] * (1.0f / DD);
    __syncthreads();
    float d = x - mu;
    red[t] = d * d;
    __syncthreads();
    for (int off = 64; off > 0; off >>= 1) {
        if (t < off) red[t] += red[t + off];
        __syncthreads();
    }
    float r = rsqrtf(red[0] * (1.0f / DD) + 1e-5f);
    out[row * DD + t] = d * r * g[t] + be[t];
}

// ---------------------------------------------------------------------------
extern "C" void kernel_launch(void* const* d_in, const int* in_sizes, int n_in,
                              void* d_out, int out_size, void* d_ws, size_t ws_size,
                              hipStream_t stream) {
    (void)in_sizes; (void)n_in; (void)out_size; (void)ws_size;
    const float* X   = (const float*)d_in[0];
    const float* Y   = (const float*)d_in[1];
    const int*   msk = (const int*)d_in[2];
    const float* W1y = (const float*)d_in[3];
    const float* W1x = (const float*)d_in[4];
    const float* b1  = (const float*)d_in[5];
    const float* W2  = (const float*)d_in[6];
    const float* b2  = (const float*)d_in[7];
    const float* Wf1 = (const float*)d_in[8];
    const float* bf1 = (const float*)d_in[9];
    const float* Wf2 = (const float*)d_in[10];
    const float* bf2 = (const float*)d_in[11];
    const float* g0  = (const float*)d_in[12];
    const float* be0 = (const float*)d_in[13];
    const float* g1  = (const float*)d_in[14];
    const float* be1 = (const float*)d_in[15];

    float* ws    = (float*)d_ws;
    float* maskT = ws;                        // B*M*N           = 524288
    float* hy    = maskT + BB * MM * NN;      // ROWS*H          = 262144
    float* hxb   = hy + ROWS * HH;            // ROWS*H          = 262144
    float* Ppart = hxb + ROWS * HH;           // MCHUNKS*B*H*N   = 2097152
    float* PT    = Ppart + MCHUNKS * BB * HH * NN;  // H*ROWS    = 262144
    float* cnt   = PT + HH * ROWS;            // ROWS            = 1024
    float* Zpre  = cnt + ROWS;                // ROWS*D          = 131072
    float* Z     = Zpre + ROWS * DD;          // ROWS*D          = 131072
    float* U     = hy;                        // reuse (hy dead after pool)
    float* Z2    = hxb;                       // reuse (hxb dead after pool)
    float* out   = (float*)d_out;

    // mask preprocessing
    transpose_mask<<<BB * MM * NN / 256, 256, 0, stream>>>(msk, maskT);
    count_mask<<<ROWS / 256, 256, 0, stream>>>(msk, cnt);

    // hy = Y @ W1y ; hxb = X @ W1x + b1          [1024x128]@[128x256]
    wmma_gemm_f32<0, false><<<128, 256, 0, stream>>>(Y, DD, W1y, HH, hy, HH, ROWS, DD,
                                                     nullptr, nullptr, nullptr);
    wmma_gemm_f32<1, false><<<128, 256, 0, stream>>>(X, DD, W1x, HH, hxb, HH, ROWS, DD,
                                                     b1, nullptr, nullptr);

    // masked relu pool over M (partials), then reduce+transpose -> PT[k][row]
    pool_kernel<<<dim3(NN / 32, MCHUNKS, BB), 256, 0, stream>>>(hy, hxb, maskT, Ppart);
    reduce_P<<<HH * ROWS / 256, 256, 0, stream>>>(Ppart, PT);

    // Zpre = X + P @ W2 + cnt*b2                 [1024x256]@[256x128]
    wmma_gemm_f32<4, true><<<64, 256, 0, stream>>>(PT, ROWS, W2, DD, Zpre, DD, ROWS, HH,
                                                   b2, X, cnt);
    layernorm_kernel<<<ROWS, 128, 0, stream>>>(Zpre, g0, be0, Z);

    // residual MLP: U = relu(Z@Wf1+bf1) ; Z2 = Z + U@Wf2 + bf2
    wmma_gemm_f32<2, false><<<128, 256, 0, stream>>>(Z, DD, Wf1, HH, U, HH, ROWS, DD,
                                                     bf1, nullptr, nullptr);
    wmma_gemm_f32<3, false><<<64, 256, 0, stream>>>(U, HH, Wf2, DD, Z2, DD, ROWS, HH,
                                                    bf2, Z, nullptr);
    layernorm_kernel<<<ROWS, 128, 0, stream>>>(Z2, g1, be1, out);
}